// sqgkt_53927609368959
// MI455X (gfx1250) — compile-verified
//
#include <hip/hip_runtime.h>

#define TT 50
#define DD 100
#define DP 112
#define NBRS 6
#define NEGV (-1000000000.0f)

typedef float v2f __attribute__((ext_vector_type(2)));
typedef float v8f __attribute__((ext_vector_type(8)));

// ---------------- LDS layout (floats) ----------------
constexpr int OFF_SH  = 0;                  // 50*DP  state history
constexpr int OFF_E0  = OFF_SH  + TT*DP;    // 2*DP   hop0 (q,u)
constexpr int OFF_E1  = OFF_E0  + 2*DP;     // 12*DP  hop1 (q rows 0-5, u rows 6-11)
constexpr int OFF_E2  = OFF_E1  + 12*DP;    // 72*DP  hop2 (q rows 0-35, u rows 36-71)
constexpr int OFF_SB  = OFF_E2  + 72*DP;    // 86*DP  staging: rows 0-71 s2, 72-83 s1, 84-85 s0
constexpr int OFF_VQ  = OFF_SB  + 86*DP;    // DP     W_query^T . wA
constexpr int OFF_VK  = OFF_VQ  + DP;       // DP     W_key^T . wB
constexpr int OFF_QS  = OFF_VK  + DP;       // 5*DP   qs_concat
constexpr int OFF_ST  = OFF_QS  + 5*DP;     // 11*DP  states
constexpr int OFF_H   = OFF_ST  + 11*DP;    // DP
constexpr int OFF_C   = OFF_H   + DP;       // DP
constexpr int OFF_G   = OFF_C   + DP;       // 400    gates
constexpr int OFF_CAT = OFF_G   + 4*DD;     // 2*DP   concat row (fusion/gates input)
constexpr int OFF_ET  = OFF_CAT + 2*DP;     // DP     e_t
constexpr int OFF_SC  = OFF_ET  + DP;       // 64     scores[50]
constexpr int OFF_QL  = OFF_SC  + 64;       // 16     ql[5], kl[11]
constexpr int OFF_LG  = OFF_QL  + 16;       // 56     logits[55]
constexpr int OFF_GG  = OFF_LG  + 56;       // 56     g[55]
constexpr int OFF_RED = OFF_GG  + 56;       // 8      scalars (bqA,bkA)
constexpr int F_TOT   = OFF_RED + 8;
// int region (after floats)
constexpr int SI_N1   = 0;    // 12
constexpr int SI_N2   = 12;   // 72
constexpr int SI_QS   = 84;   // 4
constexpr int SI_IDX  = 88;   // 10
constexpr int SI_SV   = 98;   // 11
constexpr int SI_Q    = 109;
constexpr int SI_U    = 110;
constexpr int SI_R    = 111;
constexpr int SI_M    = 112;
constexpr int SI_QN   = 113;
constexpr int SI_N3   = 128;  // 72*6 = 432 hop3 indices
constexpr int I_TOT   = SI_N3 + 72*NBRS;
constexpr size_t SMEM_BYTES = (size_t)(F_TOT + I_TOT) * 4;

__device__ __forceinline__ float sigmf(float x) { return 1.0f / (1.0f + __expf(-x)); }

// OUT[r, n] = epi( sum_k A[r*DP + k] * W[n*LDW + k] + bias[n] ),  r<R, n<N.
// All loads are UNCONDITIONAL with clamped row/col indices: out-of-range A rows /
// B columns only pollute output rows/cols that the store guard discards. This keeps
// EXEC all-ones (WMMA requirement) and avoids save/restore-exec predication.
// EPI: 0=none, 1=tanh, 2=relu. Caller must __syncthreads() after.
template <int R, int N, int K, int LDW, int EPI>
__device__ void gemm_wmma(const float* __restrict__ A,
                          const float* __restrict__ W,
                          const float* __restrict__ bias,
                          float* __restrict__ OUT, int ldo)
{
  const int lane = threadIdx.x & 31;
  const int wid  = threadIdx.x >> 5;
  const int nw   = blockDim.x >> 5;
  const int half = lane >> 4;        // which K-pair half
  const int l16  = lane & 15;
  constexpr int MT = (R + 15) >> 4;
  constexpr int NT = (N + 15) >> 4;
  for (int tile = wid; tile < MT * NT; tile += nw) {
    const int m0 = (tile % MT) << 4;
    const int n0 = (tile / MT) << 4;
    const int ar = m0 + l16;
    const int bn = n0 + l16;
    const int arc = (ar < R) ? ar : (R - 1);   // clamp: stay in bounds
    const int bnc = (bn < N) ? bn : (N - 1);
    const float* arow = A + (size_t)arc * DP;
    const float* wrow = W + (size_t)bnc * LDW;
    v8f acc = {0.f,0.f,0.f,0.f,0.f,0.f,0.f,0.f};
#pragma unroll 5
    for (int k0 = 0; k0 < K; k0 += 4) {
      const int kb = k0 + half * 2;
      const v2f a  = *(const v2f*)(arow + kb);   // ds_load_b64
      const v2f bb = *(const v2f*)(wrow + kb);   // global_load_b64
      acc = __builtin_amdgcn_wmma_f32_16x16x4_f32(false, a, false, bb, (short)0, acc, false, false);
    }
    const float bv = bias[bnc];
#pragma unroll
    for (int r = 0; r < 8; ++r) {
      const int row = m0 + r + half * 8;   // C/D layout: VGPR r -> M=r (lanes<16), M=r+8 (lanes>=16)
      if (row < R && bn < N) {
        float v = acc[r] + bv;
        if (EPI == 1)      v = tanhf(v);
        else if (EPI == 2) v = fmaxf(v, 0.f);
        OUT[(size_t)row * ldo + bn] = v;
      }
    }
  }
}

// gates = [e_t | h] @ [W1 ; W2]^T + b1 + b2   (R=1, K=2*DD split at DD, N=4*DD)
// W1/W2 select is uniform per unrolled k0 (k0<DD <=> kb<DD since k0 % 4 == 0, DD even).
__device__ void gemm_wmma_gates(const float* __restrict__ A,
                                const float* __restrict__ W1, const float* __restrict__ W2,
                                const float* __restrict__ b1, const float* __restrict__ b2,
                                float* __restrict__ OUT)
{
  const int lane = threadIdx.x & 31;
  const int wid  = threadIdx.x >> 5;
  const int nw   = blockDim.x >> 5;
  const int half = lane >> 4;
  const int l16  = lane & 15;
  constexpr int NT = (4 * DD) / 16;   // 25 (exact)
  for (int tile = wid; tile < NT; tile += nw) {
    const int bn = (tile << 4) + l16;               // always < 4*DD
    const float* w1 = W1 + (size_t)bn * DD;
    const float* w2 = W2 + (size_t)bn * DD;
    v8f acc = {0.f,0.f,0.f,0.f,0.f,0.f,0.f,0.f};
#pragma unroll 5
    for (int k0 = 0; k0 < 2 * DD; k0 += 4) {
      const int kb = k0 + half * 2;
      const v2f a = *(const v2f*)(A + kb);          // same row for all lanes; rows>0 discarded
      v2f bb;
      if (k0 < DD) bb = *(const v2f*)(w1 + kb);
      else         bb = *(const v2f*)(w2 + (kb - DD));
      acc = __builtin_amdgcn_wmma_f32_16x16x4_f32(false, a, false, bb, (short)0, acc, false, false);
    }
    if (half == 0) {                                // row 0 only (r=0, lanes<16)
      OUT[bn] = acc[0] + b1[bn] + b2[bn];
    }
  }
}

__device__ __forceinline__ void stage_s1(float* sm) {   // rows 72..83: mean(e2 group) + e1
  for (int idx = threadIdx.x; idx < 12 * DD; idx += blockDim.x) {
    int i = idx / DD, k = idx % DD;
    float acc = 0.f;
    for (int j = 0; j < NBRS; ++j) acc += sm[OFF_E2 + (i * 6 + j) * DP + k];
    sm[OFF_SB + (72 + i) * DP + k] = acc * (1.f / 6.f) + sm[OFF_E1 + i * DP + k];
  }
}

__device__ __forceinline__ void stage_s0(float* sm) {   // rows 84..85: mean(e1 side) + e0
  for (int idx = threadIdx.x; idx < 2 * DD; idx += blockDim.x) {
    int i = idx / DD, k = idx % DD;
    float acc = 0.f;
    for (int j = 0; j < NBRS; ++j) acc += sm[OFF_E1 + (i * 6 + j) * DP + k];
    sm[OFF_SB + (84 + i) * DP + k] = acc * (1.f / 6.f) + sm[OFF_E0 + i * DP + k];
  }
}

__global__ __launch_bounds__(256) void sqgkt_kernel(
    const int* __restrict__ user, const int* __restrict__ question,
    const int* __restrict__ response, const int* __restrict__ maskp,
    const int* __restrict__ q_neighbors, const int* __restrict__ s_neighbors,
    const int* __restrict__ u_neighbors, const int* __restrict__ q_neighbors_2,
    const int* __restrict__ qs_index,
    const float* __restrict__ emb_question, const float* __restrict__ emb_question_2,
    const float* __restrict__ emb_skill, const float* __restrict__ emb_user,
    const float* __restrict__ emb_response,
    const float* __restrict__ w1_q, const float* __restrict__ w2_q,
    const float* __restrict__ W_ih, const float* __restrict__ W_hh,
    const float* __restrict__ b_ih, const float* __restrict__ b_hh,
    const float* __restrict__ agg_w, const float* __restrict__ agg_b,
    const float* __restrict__ W_agg_last, const float* __restrict__ b_agg_last,
    const float* __restrict__ W_query, const float* __restrict__ b_query,
    const float* __restrict__ W_key, const float* __restrict__ b_key,
    const float* __restrict__ W_w, const float* __restrict__ b_w,
    const float* __restrict__ W_fusion, const float* __restrict__ b_fusion,
    float* __restrict__ out)
{
  extern __shared__ float sm[];
  int* si = (int*)(sm + F_TOT);
  const int b   = blockIdx.x;
  const int tid = threadIdx.x;
  const float w1 = w1_q[0], w2 = w2_q[0];

  // ---- init: zero sh/h/c, precompute vq/vk, bqA/bkA, out[b][0]=0.5 ----
  for (int i = tid; i < TT * DP; i += blockDim.x) sm[OFF_SH + i] = 0.f;
  for (int i = tid; i < DP; i += blockDim.x) { sm[OFF_H + i] = 0.f; sm[OFF_C + i] = 0.f; }
  for (int k = tid; k < DD; k += blockDim.x) {
    float aq = 0.f, ak = 0.f;
    for (int n = 0; n < DD; ++n) {
      aq += W_query[n * DD + k] * W_w[n];
      ak += W_key[n * DD + k]   * W_w[DD + n];
    }
    sm[OFF_VQ + k] = aq; sm[OFF_VK + k] = ak;
  }
  if (tid == 0) {
    float s1 = 0.f, s2 = 0.f;
    for (int n = 0; n < DD; ++n) { s1 += b_query[n] * W_w[n]; s2 += b_key[n] * W_w[DD + n]; }
    sm[OFF_RED + 0] = s1; sm[OFF_RED + 1] = s2;
    out[(size_t)b * TT + 0] = 0.5f;
  }
  __syncthreads();
  const float bqA = sm[OFF_RED + 0], bkA = sm[OFF_RED + 1];

  for (int t = 0; t < TT - 1; ++t) {
    if (tid == 0) {
      si[SI_Q]  = question[b * TT + t];
      si[SI_U]  = user[b * TT + t];
      si[SI_R]  = response[b * TT + t];
      si[SI_M]  = maskp[b * TT + t];
      si[SI_QN] = question[b * TT + t + 1];
    }
    __syncthreads();
    const int qt = si[SI_Q], ut = si[SI_U], rt = si[SI_R], mm = si[SI_M], qn = si[SI_QN];

    if (mm == 1) {
      // ---- gather indices ----
      if (tid < 12) {
        si[SI_N1 + tid] = (tid < 6) ? q_neighbors[qt * NBRS + tid]
                                    : u_neighbors[ut * NBRS + (tid - 6)];
      }
      for (int k = tid; k < DD; k += blockDim.x) {
        sm[OFF_E0 + k]      = emb_question[(size_t)qt * DD + k];
        sm[OFF_E0 + DP + k] = emb_user[(size_t)ut * DD + k];
      }
      __syncthreads();
      if (tid < 72) {
        int i = tid / 6, j = tid % 6, n1 = si[SI_N1 + i];
        si[SI_N2 + tid] = (i < 6) ? s_neighbors[n1 * NBRS + j] : q_neighbors_2[n1 * NBRS + j];
      }
      for (int idx = tid; idx < 12 * DD; idx += blockDim.x) {
        int r = idx / DD, k = idx % DD, n1 = si[SI_N1 + r];
        sm[OFF_E1 + r * DP + k] = (r < 6) ? emb_skill[(size_t)n1 * DD + k]
                                          : emb_question_2[(size_t)n1 * DD + k];
      }
      __syncthreads();
      if (tid < 72 * NBRS) {
        int r = tid / NBRS, j = tid % NBRS, n2 = si[SI_N2 + r];
        si[SI_N3 + tid] = (r < 36) ? q_neighbors[n2 * NBRS + j] : u_neighbors[n2 * NBRS + j];
      }
      for (int idx = tid; idx < 72 * DD; idx += blockDim.x) {
        int r = idx / DD, k = idx % DD, n2 = si[SI_N2 + r];
        sm[OFF_E2 + r * DP + k] = (r < 36) ? emb_question[(size_t)n2 * DD + k]
                                           : emb_user[(size_t)n2 * DD + k];
      }
      __syncthreads();
      // ---- round 0: stage s2 (hop3 mean + e2), s1, s0; then 3 WMMA gemms ----
      for (int idx = tid; idx < 72 * DD; idx += blockDim.x) {
        int r = idx / DD, k = idx % DD;
        float acc = 0.f;
        if (r < 36) { for (int j = 0; j < NBRS; ++j) acc += emb_skill[(size_t)si[SI_N3 + r * NBRS + j] * DD + k]; }
        else        { for (int j = 0; j < NBRS; ++j) acc += emb_question_2[(size_t)si[SI_N3 + r * NBRS + j] * DD + k]; }
        sm[OFF_SB + r * DP + k] = acc * (1.f / 6.f) + sm[OFF_E2 + r * DP + k];
      }
      stage_s1(sm);
      stage_s0(sm);
      __syncthreads();
      gemm_wmma< 2, DD, DD, DD, 1>(sm + OFF_SB + 84 * DP, agg_w,               agg_b,          sm + OFF_E0, DP);
      gemm_wmma<12, DD, DD, DD, 1>(sm + OFF_SB + 72 * DP, agg_w + DD * DD,     agg_b + DD,     sm + OFF_E1, DP);
      gemm_wmma<72, DD, DD, DD, 1>(sm + OFF_SB,           agg_w + 2 * DD * DD, agg_b + 2 * DD, sm + OFF_E2, DP);
      __syncthreads();
      // ---- round 1 ----
      stage_s1(sm);
      stage_s0(sm);
      __syncthreads();
      gemm_wmma< 2, DD, DD, DD, 1>(sm + OFF_SB + 84 * DP, agg_w,           agg_b,      sm + OFF_E0, DP);
      gemm_wmma<12, DD, DD, DD, 1>(sm + OFF_SB + 72 * DP, agg_w + DD * DD, agg_b + DD, sm + OFF_E1, DP);
      __syncthreads();
      // ---- round 2 ----
      stage_s0(sm);
      __syncthreads();
      gemm_wmma< 2, DD, DD, DD, 1>(sm + OFF_SB + 84 * DP, agg_w, agg_b, sm + OFF_E0, DP);
      __syncthreads();
      // ---- final aggregate projection ----
      for (int idx = tid; idx < 2 * DD; idx += blockDim.x) {
        int i = idx / DD, k = idx % DD;
        sm[OFF_SB + (84 + i) * DP + k] = sm[OFF_E0 + i * DP + k];
      }
      __syncthreads();
      gemm_wmma< 2, DD, DD, DD, 1>(sm + OFF_SB + 84 * DP, W_agg_last, b_agg_last, sm + OFF_E0, DP);
      __syncthreads();
    }

    // ---- e_hat, fusion ----
    for (int k = tid; k < DD; k += blockDim.x) {
      float eh = (mm == 1)
        ? (w1 * sm[OFF_E0 + k] + w2 * sm[OFF_E0 + DP + k])
        : (w1 * emb_question[(size_t)qt * DD + k] + w2 * emb_question_2[(size_t)qt * DD + k]);
      sm[OFF_CAT + k]      = eh;
      sm[OFF_CAT + DD + k] = emb_response[(size_t)rt * DD + k];
    }
    __syncthreads();
    gemm_wmma<1, DD, 2 * DD, 2 * DD, 2>(sm + OFF_CAT, W_fusion, b_fusion, sm + OFF_ET, DP);
    __syncthreads();

    // ---- LSTM gates ----
    for (int k = tid; k < DD; k += blockDim.x) {
      sm[OFF_CAT + k]      = sm[OFF_ET + k];
      sm[OFF_CAT + DD + k] = sm[OFF_H + k];
    }
    __syncthreads();
    gemm_wmma_gates(sm + OFF_CAT, W_ih, W_hh, b_ih, b_hh, sm + OFF_G);
    __syncthreads();
    for (int k = tid; k < DD; k += blockDim.x) {
      float gi = sm[OFF_G + k], gf = sm[OFF_G + DD + k];
      float gg = sm[OFF_G + 2 * DD + k], go = sm[OFF_G + 3 * DD + k];
      float c2 = sigmf(gf) * sm[OFF_C + k] + sigmf(gi) * tanhf(gg);
      float h2 = sigmf(go) * tanhf(c2);
      if (mm == 1) { sm[OFF_H + k] = h2; sm[OFF_C + k] = c2; }
      sm[OFF_SH + t * DP + k] = sm[OFF_H + k];
    }
    __syncthreads();

    // ---- prediction ----
    if (tid < 4) si[SI_QS + tid] = qs_index[qn * 4 + tid];
    __syncthreads();
    for (int idx = tid; idx < 5 * DD; idx += blockDim.x) {
      int r = idx / DD, k = idx % DD;
      sm[OFF_QS + r * DP + k] = (r == 0) ? emb_question[(size_t)qn * DD + k]
                                         : emb_skill[(size_t)si[SI_QS + r - 1] * DD + k];
    }
    __syncthreads();
    if (tid < TT) {
      float s = 0.f;
      int qq = question[b * TT + tid];
      for (int k = 0; k < DD; ++k) s += emb_question[(size_t)qq * DD + k] * sm[OFF_QS + k];
      sm[OFF_SC + tid] = (tid < t) ? s : NEGV;
    }
    __syncthreads();
    if (tid == 0) {
      for (int r = 0; r < 10; ++r) {       // serial top-10 (stable: first occurrence of max)
        float mv = -3.0e38f; int mi = 0;
        for (int j = 0; j < TT; ++j) { float v = sm[OFF_SC + j]; if (v > mv) { mv = v; mi = j; } }
        si[SI_IDX + r] = mi;
        si[SI_SV + 1 + r] = (mi < t) ? 1 : 0;
        sm[OFF_SC + mi] = -3.4e38f;
      }
      si[SI_SV + 0] = 1;
    }
    __syncthreads();
    for (int idx = tid; idx < 11 * DD; idx += blockDim.x) {
      int r = idx / DD, k = idx % DD;
      sm[OFF_ST + r * DP + k] = (r == 0) ? sm[OFF_H + k]
                                         : sm[OFF_SH + si[SI_IDX + r - 1] * DP + k];
    }
    __syncthreads();
    if (tid < 16) {
      float s = 0.f;
      if (tid < 5) { for (int k = 0; k < DD; ++k) s += sm[OFF_QS + tid * DP + k] * sm[OFF_VQ + k]; s += bqA; }
      else { int r = tid - 5; for (int k = 0; k < DD; ++k) s += sm[OFF_ST + r * DP + k] * sm[OFF_VK + k]; s += bkA; }
      sm[OFF_QL + tid] = s;
    }
    __syncthreads();
    if (tid < 55) {
      int q2 = tid / 11, s2 = tid % 11;
      float lg = sm[OFF_QL + q2] + sm[OFF_QL + 5 + s2] + b_w[0];
      if (!si[SI_SV + s2]) lg = NEGV;
      sm[OFF_LG + tid] = lg;
      float g = 0.f;
      for (int k = 0; k < DD; ++k) g += sm[OFF_QS + q2 * DP + k] * sm[OFF_ST + s2 * DP + k];
      sm[OFF_GG + tid] = g;
    }
    __syncthreads();
    if (tid == 0) {
      float mx = -3.0e38f;
      for (int i = 0; i < 55; ++i) mx = fmaxf(mx, sm[OFF_LG + i]);
      float z = 0.f, acc = 0.f;
      for (int i = 0; i < 55; ++i) {
        float e = __expf(sm[OFF_LG + i] - mx);
        z += e; acc += e * sm[OFF_GG + i];
      }
      out[(size_t)b * TT + t + 1] = sigmf(acc / z);
    }
    __syncthreads();
  }
}

extern "C" void kernel_launch(void* const* d_in, const int* in_sizes, int n_in,
                              void* d_out, int out_size, void* d_ws, size_t ws_size,
                              hipStream_t stream) {
  (void)n_in; (void)out_size; (void)d_ws; (void)ws_size;
  const int B = in_sizes[1] / TT;
  hipFuncSetAttribute(reinterpret_cast<const void*>(sqgkt_kernel),
                      hipFuncAttributeMaxDynamicSharedMemorySize, (int)SMEM_BYTES);
  sqgkt_kernel<<<dim3(B), dim3(256), SMEM_BYTES, stream>>>(
      (const int*)d_in[0], (const int*)d_in[1], (const int*)d_in[2], (const int*)d_in[3],
      (const int*)d_in[4], (const int*)d_in[5], (const int*)d_in[6], (const int*)d_in[7],
      (const int*)d_in[8],
      (const float*)d_in[9], (const float*)d_in[10], (const float*)d_in[11],
      (const float*)d_in[12], (const float*)d_in[13],
      (const float*)d_in[14], (const float*)d_in[15],
      (const float*)d_in[16], (const float*)d_in[17], (const float*)d_in[18], (const float*)d_in[19],
      (const float*)d_in[20], (const float*)d_in[21], (const float*)d_in[22], (const float*)d_in[23],
      (const float*)d_in[24], (const float*)d_in[25], (const float*)d_in[26], (const float*)d_in[27],
      (const float*)d_in[28], (const float*)d_in[29], (const float*)d_in[30], (const float*)d_in[31],
      (float*)d_out);
}